// SmallWorldAttention_84078279786896
// MI455X (gfx1250) — compile-verified
//
#include <hip/hip_runtime.h>
#include <hip/hip_bf16.h>

// ---------------------------------------------------------------------------
// Problem constants (match reference)
// ---------------------------------------------------------------------------
#define L_SEQ   4096
#define QDIM    1024
#define NHEAD   16
#define HDIM    64
#define INNER   1024          // NHEAD * HDIM
#define SCALE   0.125f        // 1/sqrt(64)
#define RMS_EPS 1e-6f

typedef float v2f __attribute__((ext_vector_type(2)));
typedef float v8f __attribute__((ext_vector_type(8)));
typedef int   swa_v4i __attribute__((vector_size(16)));   // int4 for async-LDS builtin

// --- CDNA5 async global->LDS copy support (guarded; falls back cleanly) ---
#if defined(__has_builtin)
#if __has_builtin(__builtin_amdgcn_global_load_async_to_lds_b128)
#define SWA_HAVE_ASYNC_LDS 1
#endif
#endif

// Builtin expects plain v4i* (generic) pointers on this toolchain.
#define SWA_GPTR(p) ((swa_v4i*)(p))
#define SWA_LPTR(p) ((swa_v4i*)(p))

__device__ __forceinline__ void swa_wait_async() {
#if defined(__has_builtin)
#if __has_builtin(__builtin_amdgcn_s_wait_asynccnt)
    __builtin_amdgcn_s_wait_asynccnt(0);
#else
    asm volatile("s_wait_asynccnt 0x0" ::: "memory");
#endif
#else
    asm volatile("s_wait_asynccnt 0x0" ::: "memory");
#endif
}

// ---------------------------------------------------------------------------
// WMMA fp32 GEMM:  C[M,N] = A[M,K] @ W[N,K]^T + bias[N]
// Block tile 64(M) x 128(N), 256 threads = 8 waves arranged 2x4,
// each wave owns a 32x32 tile = 2x2 subtiles of 16x16 (v8f accumulators).
// A tile staged with GLOBAL_LOAD_ASYNC_TO_LDS_B128 (ASYNCcnt tracked);
// W tile staged through registers (needs K-major transpose for the B frag).
// 4 x v_wmma_f32_16x16x4_f32 per k-step.
// ---------------------------------------------------------------------------
#define GEMM_TM 64
#define GEMM_TN 128
#define GEMM_TK 16

__global__ __launch_bounds__(256)
void swa_gemm_xWT_bias(const float* __restrict__ A,
                       const float* __restrict__ W,
                       const float* __restrict__ bias,
                       float* __restrict__ C,
                       int M, int N, int K)
{
    __shared__ float As[GEMM_TM][GEMM_TK];       // 64 x 16  (4 KB)
    __shared__ float Bs[GEMM_TK][GEMM_TN + 4];   // 16 x 132 (padded, 8.25 KB)

    const int tid  = threadIdx.x;
    const int lane = tid & 31;
    const int wave = tid >> 5;
    const int wm   = (wave >> 2) * 32;           // 0 or 32
    const int wn   = (wave & 3) * 32;            // 0,32,64,96
    const int m0   = blockIdx.y * GEMM_TM;
    const int n0   = blockIdx.x * GEMM_TN;

    v8f acc[2][2] = {};                          // zero-init accumulators

    const int arow = tid >> 2;                   // 0..63
    const int aseg = (tid & 3) * 4;              // 0,4,8,12

    for (int k0 = 0; k0 < K; k0 += GEMM_TK) {
        // --- stage A tile: one 16B async copy per thread, no VGPR bounce ---
        const float* asrc = &A[(size_t)(m0 + arow) * K + k0 + aseg];
#if defined(SWA_HAVE_ASYNC_LDS)
        __builtin_amdgcn_global_load_async_to_lds_b128(
            SWA_GPTR(asrc), SWA_LPTR(&As[arow][aseg]), 0, 0);
#else
        *reinterpret_cast<float4*>(&As[arow][aseg]) =
            *reinterpret_cast<const float4*>(asrc);
#endif

        // --- stage W tile, transposed into Bs[k][n] (B is KxN for WMMA) ---
        #pragma unroll
        for (int r = 0; r < 2; ++r) {
            const int n = arow + r * 64;
            float4 w4 = *reinterpret_cast<const float4*>(
                &W[(size_t)(n0 + n) * K + k0 + aseg]);
            Bs[aseg + 0][n] = w4.x;
            Bs[aseg + 1][n] = w4.y;
            Bs[aseg + 2][n] = w4.z;
            Bs[aseg + 3][n] = w4.w;
        }
#if defined(SWA_HAVE_ASYNC_LDS)
        swa_wait_async();                        // A-tile bytes landed in LDS
#endif
        __syncthreads();

        // --- 4 WMMA k-steps of depth 4 ---
        #pragma unroll
        for (int kk = 0; kk < GEMM_TK; kk += 4) {
            // A frag layout (16x4 f32): lanes 0-15 hold K=0,1; lanes 16-31 K=2,3
            const int klo = kk + ((lane >> 4) << 1);
            const int mn  = lane & 15;
            v2f afrag[2], bfrag[2];
            #pragma unroll
            for (int mi = 0; mi < 2; ++mi) {
                afrag[mi].x = As[wm + mi * 16 + mn][klo + 0];
                afrag[mi].y = As[wm + mi * 16 + mn][klo + 1];
            }
            #pragma unroll
            for (int nj = 0; nj < 2; ++nj) {
                bfrag[nj].x = Bs[klo + 0][wn + nj * 16 + mn];
                bfrag[nj].y = Bs[klo + 1][wn + nj * 16 + mn];
            }
            #pragma unroll
            for (int mi = 0; mi < 2; ++mi)
                #pragma unroll
                for (int nj = 0; nj < 2; ++nj)
                    acc[mi][nj] = __builtin_amdgcn_wmma_f32_16x16x4_f32(
                        false, afrag[mi], false, bfrag[nj],
                        (short)0, acc[mi][nj], false, false);
        }
        __syncthreads();
    }

    // --- epilogue: C/D layout: VGPR r, lanes 0-15 -> M=r, lanes 16-31 -> M=r+8
    const int crow = (lane >> 4) * 8;
    const int ccol = lane & 15;
    #pragma unroll
    for (int mi = 0; mi < 2; ++mi)
        #pragma unroll
        for (int nj = 0; nj < 2; ++nj) {
            const int n = n0 + wn + nj * 16 + ccol;
            const float bn = bias[n];
            #pragma unroll
            for (int r = 0; r < 8; ++r) {
                const int m = m0 + wm + mi * 16 + crow + r;
                C[(size_t)m * N + n] = acc[mi][nj][r] + bn;
            }
        }
}

// ---------------------------------------------------------------------------
// In-place RMS norm over the last (1024) axis: y = y * rsqrt(mean(y^2)+eps) * w
// one block (256 threads) per row; wave32 butterfly + 8-slot LDS reduction
// ---------------------------------------------------------------------------
__global__ __launch_bounds__(256)
void swa_rmsnorm_inplace(float* __restrict__ y, const float* __restrict__ w)
{
    __shared__ float red[8];
    const int row = blockIdx.x;
    const int tid = threadIdx.x;

    float4 vv = *reinterpret_cast<float4*>(&y[(size_t)row * INNER + tid * 4]);
    float ss = vv.x * vv.x + vv.y * vv.y + vv.z * vv.z + vv.w * vv.w;
    #pragma unroll
    for (int off = 16; off >= 1; off >>= 1)
        ss += __shfl_xor(ss, off, 32);
    if ((tid & 31) == 0) red[tid >> 5] = ss;
    __syncthreads();
    const float tot = red[0] + red[1] + red[2] + red[3] +
                      red[4] + red[5] + red[6] + red[7];
    const float scale = rsqrtf(tot * (1.0f / (float)INNER) + RMS_EPS);

    float4 w4 = *reinterpret_cast<const float4*>(&w[tid * 4]);
    vv.x *= scale * w4.x;
    vv.y *= scale * w4.y;
    vv.z *= scale * w4.z;
    vv.w *= scale * w4.w;
    *reinterpret_cast<float4*>(&y[(size_t)row * INNER + tid * 4]) = vv;
}

// ---------------------------------------------------------------------------
// Shifted (small-world) attention.
// One block per sequence position l (512 threads = 16 waves = 16 heads).
// Each lane owns 2 of the 64 head dims. Scores parked lane-per-shift,
// softmax fully in registers via wave32 butterflies.
// ---------------------------------------------------------------------------
__global__ __launch_bounds__(512)
void swa_attention(const float* __restrict__ q,
                   const float* __restrict__ k,
                   const float* __restrict__ v,
                   const int*   __restrict__ shifts,
                   const float* __restrict__ edge_bias,
                   float* __restrict__ out, int S)
{
    const int l    = blockIdx.x;
    const int h    = threadIdx.x >> 5;
    const int lane = threadIdx.x & 31;
    const size_t base = (size_t)h * HDIM + (size_t)lane * 2;

    const float2 q2 =
        *reinterpret_cast<const float2*>(&q[(size_t)l * INNER + base]);

    // lane s caches gather index for shift s  (S = 31 <= 32)
    const int myShift = (lane < S) ? shifts[lane] : 0;
    const int myIdx   = (l + myShift + L_SEQ) & (L_SEQ - 1);

    // ---- scores: q[l,h,:] . k[(l+shift_s)%L, h, :] ----
    float myScore = -3.0e38f;
    for (int s = 0; s < S; ++s) {
        const int idx = __shfl(myIdx, s, 32);
        const float2 k2 =
            *reinterpret_cast<const float2*>(&k[(size_t)idx * INNER + base]);
        float p = q2.x * k2.x + q2.y * k2.y;
        #pragma unroll
        for (int off = 16; off >= 1; off >>= 1)
            p += __shfl_xor(p, off, 32);
        if (lane == s) myScore = p;
    }

    // ---- softmax over the S shifts (lane s holds shift s) ----
    float score = (lane < S) ? (myScore * SCALE + edge_bias[h * S + lane])
                             : -3.0e38f;
    float mx = score;
    #pragma unroll
    for (int off = 16; off >= 1; off >>= 1)
        mx = fmaxf(mx, __shfl_xor(mx, off, 32));
    float e = (lane < S) ? __expf(score - mx) : 0.0f;
    float sum = e;
    #pragma unroll
    for (int off = 16; off >= 1; off >>= 1)
        sum += __shfl_xor(sum, off, 32);
    const float attn = e / sum;

    // ---- out[l,h,:] = sum_s attn_s * v[(l+shift_s)%L, h, :] ----
    float accx = 0.0f, accy = 0.0f;
    for (int s = 0; s < S; ++s) {
        const float a  = __shfl(attn, s, 32);
        const int  idx = __shfl(myIdx, s, 32);
        const float2 v2 =
            *reinterpret_cast<const float2*>(&v[(size_t)idx * INNER + base]);
        accx += a * v2.x;
        accy += a * v2.y;
    }
    float2 o; o.x = accx; o.y = accy;
    *reinterpret_cast<float2*>(&out[(size_t)l * INNER + base]) = o;
}

// ---------------------------------------------------------------------------
// Launch
// ---------------------------------------------------------------------------
extern "C" void kernel_launch(void* const* d_in, const int* in_sizes, int n_in,
                              void* d_out, int out_size, void* d_ws, size_t ws_size,
                              hipStream_t stream)
{
    const float* x    = (const float*)d_in[0];
    const float* Wq   = (const float*)d_in[1];
    const float* bq   = (const float*)d_in[2];
    const float* Wk   = (const float*)d_in[3];
    const float* bk   = (const float*)d_in[4];
    const float* Wv   = (const float*)d_in[5];
    const float* bv   = (const float*)d_in[6];
    const float* qn_w = (const float*)d_in[7];
    const float* kn_w = (const float*)d_in[8];
    const float* ebias= (const float*)d_in[9];
    const float* Wo   = (const float*)d_in[10];
    const float* bo   = (const float*)d_in[11];
    const int*   shifts = (const int*)d_in[12];
    const int S = in_sizes[12];                  // 31

    const size_t mat = (size_t)L_SEQ * INNER;    // 4M floats = 16MB
    float* qbuf = (float*)d_ws;
    float* kbuf = qbuf + mat;
    float* vbuf = kbuf + mat;
    float* abuf = vbuf + mat;
    float* outp = (float*)d_out;

    dim3 ggrid(INNER / GEMM_TN, L_SEQ / GEMM_TM);   // (8, 64)

    // q_raw, k_raw, v projections (bias fused)
    swa_gemm_xWT_bias<<<ggrid, 256, 0, stream>>>(x, Wq, bq, qbuf, L_SEQ, INNER, QDIM);
    swa_gemm_xWT_bias<<<ggrid, 256, 0, stream>>>(x, Wk, bk, kbuf, L_SEQ, INNER, QDIM);
    swa_gemm_xWT_bias<<<ggrid, 256, 0, stream>>>(x, Wv, bv, vbuf, L_SEQ, INNER, QDIM);

    // RMS norm q, k (in place, over full 1024-dim row as in reference)
    swa_rmsnorm_inplace<<<L_SEQ, 256, 0, stream>>>(qbuf, qn_w);
    swa_rmsnorm_inplace<<<L_SEQ, 256, 0, stream>>>(kbuf, kn_w);

    // shifted attention
    swa_attention<<<L_SEQ, 512, 0, stream>>>(qbuf, kbuf, vbuf, shifts, ebias, abuf, S);

    // output projection
    swa_gemm_xWT_bias<<<dim3(QDIM / GEMM_TN, L_SEQ / GEMM_TM), 256, 0, stream>>>(
        abuf, Wo, bo, outp, L_SEQ, QDIM, INNER);
}